// LogSoftmaxLoss_3762391351869
// MI455X (gfx1250) — compile-verified
//
#include <hip/hip_runtime.h>
#include <hip/hip_bf16.h>
#include <stdint.h>

// ---------------------------------------------------------------------------
// Fused contrastive log-softmax loss for MI455X (gfx1250), wave32 + WMMA.
//   S = A @ T^T / tau, symmetric fid mask (diag kept),
//   out = mean_i( lse_row(S)[i] + lse_col(S)[i] - 2*S[i,i] )
// Never materialize S; flash-style online logsumexp fused into an fp32 WMMA
// GEMM (v_wmma_f32_16x16x4_f32). Column LSE = row LSE of the swapped product
// (mask is symmetric), so the same kernel runs twice.
// Tile pipeline: double-buffered LDS fed by GLOBAL_LOAD_ASYNC_TO_LDS_B128
// (ASYNCcnt), overlapping next-tile DMA with the current tile's WMMA chain.
// ---------------------------------------------------------------------------

typedef float v2f __attribute__((ext_vector_type(2)));
typedef float v4f __attribute__((ext_vector_type(4)));
typedef float v8f __attribute__((ext_vector_type(8)));

#define N_SAMPLES   8192
#define EDIM        128
#define INV_T       (1.0f / 0.07f)
#define CSPLIT      8               // column chunks (grid.y)
#define ROWS_PER_WG 128             // 8 waves * 16 rows
#define ROW_BLOCKS  (N_SAMPLES / ROWS_PER_WG)      // 64
#define COLS_PER_CHUNK (N_SAMPLES / CSPLIT)        // 1024
#define TILES_PER_CHUNK (COLS_PER_CHUNK / 16)      // 64
#define LDS_STRIDE  132             // 128 + 4 pad: conflict-free, keeps 16B align

__global__ __launch_bounds__(256) void lse_pass(
    const float* __restrict__ X,      // "row" embeddings  [N, E]
    const float* __restrict__ Y,      // "col" embeddings  [N, E]
    const int*   __restrict__ fids,   // [N]
    float* __restrict__ pmax,         // [N, CSPLIT]
    float* __restrict__ psum)         // [N, CSPLIT]
{
    const int tid   = threadIdx.x;
    const int wave  = tid >> 5;           // 0..7
    const int lane  = tid & 31;
    const int ln16  = lane & 15;
    const int half  = lane >> 4;          // 0 or 1
    const int chunk = blockIdx.y;

    const int strip_row0 = blockIdx.x * ROWS_PER_WG + wave * 16;
    const int chunk_col0 = chunk * COLS_PER_CHUNK;

    // Double-buffered Y tile: 2 x (16 cols x 128 k, padded) = 16.9 KB LDS.
    __shared__ float ytile[2][16 * LDS_STRIDE];

    // Async DMA of one 16-col tile straight into LDS (2 x B128 per thread).
    auto issue_tile = [&](int col0, int buf, bool pf) {
        #pragma unroll
        for (int i = 0; i < 2; ++i) {
            const int f  = tid + i * 256;       // 0..511
            const int c  = f >> 5;              // 0..15
            const int kc = f & 31;              // 0..31 (x4 floats)
            const float* gsrc = Y + (size_t)(col0 + c) * EDIM + kc * 4;
            const uint32_t ldsoff =
                (uint32_t)(uintptr_t)(const void*)&ytile[buf][c * LDS_STRIDE + kc * 4];
            asm volatile("global_load_async_to_lds_b128 %0, %1, off"
                         :: "v"(ldsoff), "v"(gsrc) : "memory");
            if (pf)                              // warm L2 for the tile after
                __builtin_prefetch(gsrc + 16 * EDIM, 0, 1);
        }
    };

    // --- A strip in registers, laid out for V_WMMA_F32_16X16X4_F32 ---
    // A (16x4 f32): lane l holds row m=l%16; slot s holds K = 2*(l/16)+s.
    v2f areg[32];
    {
        const float* ap = X + (size_t)(strip_row0 + ln16) * EDIM + 2 * half;
        #pragma unroll
        for (int kk = 0; kk < 32; ++kk)
            areg[kk] = *(const v2f*)(ap + 4 * kk);
    }

    // Row fids for the 8 C-matrix slots this lane owns (row = base + v + 8*half).
    int fidr[8];
    #pragma unroll
    for (int v = 0; v < 8; ++v)
        fidr[v] = fids[strip_row0 + v + 8 * half];

    // Online logsumexp state per owned element stream.
    float m[8], s[8];
    #pragma unroll
    for (int v = 0; v < 8; ++v) { m[v] = -3.0e38f; s[v] = 0.0f; }

    issue_tile(chunk_col0, 0, true);          // prologue: tile 0 in flight

    for (int ct = 0; ct < TILES_PER_CHUNK; ++ct) {
        const int buf  = ct & 1;
        const int col0 = chunk_col0 + ct * 16;

        if (ct + 1 < TILES_PER_CHUNK) {
            // Overlap: DMA tile ct+1 into the other buffer (its last readers
            // passed the end-of-iteration barrier of ct-1 already).
            issue_tile(col0 + 16, buf ^ 1, ct + 2 < TILES_PER_CHUNK);
            // <=2 outstanding  ==>  tile ct's 2 async ops have landed.
            asm volatile("s_wait_asynccnt 0x2" ::: "memory");
        } else {
            asm volatile("s_wait_asynccnt 0x0" ::: "memory");
        }
        __syncthreads();                      // all waves' tile-ct data in LDS

        const int   c  = col0 + ln16;
        const int   fc = fids[c];
        const float* bp = &ytile[buf][ln16 * LDS_STRIDE + 2 * half];

        // Preload all B fragments so the WMMA chain runs without DS stalls.
        v2f breg[32];
        #pragma unroll
        for (int kk = 0; kk < 32; ++kk)
            breg[kk] = *(const v2f*)(bp + 4 * kk);

        // Two accumulators break the WMMA RAW chain; 32 K-steps of 4.
        v8f acc0 = {0.f,0.f,0.f,0.f,0.f,0.f,0.f,0.f};
        v8f acc1 = {0.f,0.f,0.f,0.f,0.f,0.f,0.f,0.f};
        #pragma unroll
        for (int kk = 0; kk < 32; kk += 2) {
            acc0 = __builtin_amdgcn_wmma_f32_16x16x4_f32(
                false, areg[kk],     false, breg[kk],     (short)0, acc0, false, false);
            acc1 = __builtin_amdgcn_wmma_f32_16x16x4_f32(
                false, areg[kk + 1], false, breg[kk + 1], (short)0, acc1, false, false);
        }

        // Mask + online (max,sum) update. C layout: slot v, this lane ->
        // row = strip_row0 + v + 8*half, col = c.
        #pragma unroll
        for (int v = 0; v < 8; ++v) {
            const int   r    = strip_row0 + v + 8 * half;
            float       x    = (acc0[v] + acc1[v]) * INV_T;
            const bool  keep = (fidr[v] != fc) || (r == c);
            x = keep ? x : -__builtin_inff();
            const float mo = m[v];
            const float mn = fmaxf(mo, x);
            s[v] = s[v] * __expf(mo - mn) + __expf(x - mn);
            m[v] = mn;
        }
        __syncthreads();                      // buf reusable for tile ct+2
    }

    // Merge the 16 lanes of each row (xor masks < 16 never cross the half).
    #pragma unroll
    for (int v = 0; v < 8; ++v) {
        float mv = m[v], sv = s[v];
        #pragma unroll
        for (int off = 1; off < 16; off <<= 1) {
            const float mo = __shfl_xor(mv, off, 32);
            const float so = __shfl_xor(sv, off, 32);
            const float mn = fmaxf(mv, mo);
            sv = sv * __expf(mv - mn) + so * __expf(mo - mn);
            mv = mn;
        }
        if (ln16 == 0) {
            const int r = strip_row0 + v + 8 * half;
            pmax[(size_t)r * CSPLIT + chunk] = mv;
            psum[(size_t)r * CSPLIT + chunk] = sv;
        }
    }
}

__global__ __launch_bounds__(1024) void finalize_loss(
    const float* __restrict__ A, const float* __restrict__ T,
    const float* __restrict__ pm_r, const float* __restrict__ ps_r,
    const float* __restrict__ pm_c, const float* __restrict__ ps_c,
    float* __restrict__ out)
{
    __shared__ float red[1024];
    float acc = 0.0f;

    for (int r = threadIdx.x; r < N_SAMPLES; r += 1024) {
        float m1 = -3.0e38f, s1 = 0.0f, m2 = -3.0e38f, s2 = 0.0f;
        #pragma unroll
        for (int c = 0; c < CSPLIT; ++c) {
            float mc = pm_r[(size_t)r * CSPLIT + c];
            float sc = ps_r[(size_t)r * CSPLIT + c];
            float mn = fmaxf(m1, mc);
            s1 = s1 * __expf(m1 - mn) + sc * __expf(mc - mn);
            m1 = mn;
            mc = pm_c[(size_t)r * CSPLIT + c];
            sc = ps_c[(size_t)r * CSPLIT + c];
            mn = fmaxf(m2, mc);
            s2 = s2 * __expf(m2 - mn) + sc * __expf(mc - mn);
            m2 = mn;
        }
        const float lse_r = m1 + __logf(s1);
        const float lse_c = m2 + __logf(s2);

        // diagonal S[r,r]
        const v4f* ap = (const v4f*)(A + (size_t)r * EDIM);
        const v4f* tp = (const v4f*)(T + (size_t)r * EDIM);
        float d = 0.0f;
        #pragma unroll
        for (int k = 0; k < EDIM / 4; ++k) {
            v4f a = ap[k], t = tp[k];
            d += a[0]*t[0] + a[1]*t[1] + a[2]*t[2] + a[3]*t[3];
        }
        acc += lse_r + lse_c - 2.0f * d * INV_T;
    }

    red[threadIdx.x] = acc;
    __syncthreads();
    for (int off = 512; off > 0; off >>= 1) {
        if (threadIdx.x < off) red[threadIdx.x] += red[threadIdx.x + off];
        __syncthreads();
    }
    if (threadIdx.x == 0) out[0] = red[0] / (float)N_SAMPLES;
}

extern "C" void kernel_launch(void* const* d_in, const int* in_sizes, int n_in,
                              void* d_out, int out_size, void* d_ws, size_t ws_size,
                              hipStream_t stream) {
    (void)in_sizes; (void)n_in; (void)out_size; (void)ws_size;
    const float* A    = (const float*)d_in[0];   // audio_embeds [8192,128] f32
    const float* T    = (const float*)d_in[1];   // text_embeds  [8192,128] f32
    const int*   fids = (const int*)d_in[2];     // [8192] i32

    // Workspace layout (floats): 4 * N * CSPLIT = 256K floats = 1 MB.
    float* ws   = (float*)d_ws;
    float* pm_r = ws;
    float* ps_r = ws + (size_t)N_SAMPLES * CSPLIT;
    float* pm_c = ws + (size_t)2 * N_SAMPLES * CSPLIT;
    float* ps_c = ws + (size_t)3 * N_SAMPLES * CSPLIT;

    dim3 grid(ROW_BLOCKS, CSPLIT);
    lse_pass<<<grid, 256, 0, stream>>>(A, T, fids, pm_r, ps_r);   // row LSE
    lse_pass<<<grid, 256, 0, stream>>>(T, A, fids, pm_c, ps_c);   // col LSE (S^T)
    finalize_loss<<<1, 1024, 0, stream>>>(A, T, pm_r, ps_r, pm_c, ps_c,
                                          (float*)d_out);
}